// GVAE_12163347383058
// MI455X (gfx1250) — compile-verified
//
#include <hip/hip_runtime.h>

typedef __attribute__((ext_vector_type(2))) float v2f;
typedef __attribute__((ext_vector_type(8))) float v8f;
typedef unsigned int u32;
typedef __attribute__((ext_vector_type(4))) u32 u32x4;
typedef __attribute__((ext_vector_type(4))) int i32x4;
typedef __attribute__((ext_vector_type(8))) int i32x8;

static constexpr int NN = 8192;   // nodes
static constexpr int NX = 512;    // input features
static constexpr int NH = 256;    // hidden
static constexpr int NZ = 64;     // latent
static constexpr int NE = 262144; // edges
static constexpr int NC = 2 * NZ; // concat mean|log_sigma = 128

static constexpr double N2d  = (double)NN * (double)NN;                            // 67108864
static constexpr float  POS_W = (float)((N2d - (double)NE) / (double)NE);          // 255.0
static constexpr float  NORM_LOSS = (float)(N2d / ((N2d - (double)NE) * 2.0));     // ~0.50196

// ---- workspace layout (bytes) ----
static constexpr size_t ACC_OFF    = 0;                               // 2 doubles: [0]=wce, [1]=kl
static constexpr size_t ROWPTR_OFF = 256;                             // (NN+1) ints
static constexpr size_t XW1_OFF    = 65536;                           // NN*NH f32 (8 MB)
static constexpr size_t H1_OFF     = XW1_OFF  + (size_t)NN * NH * 4;  // 8 MB
static constexpr size_t HW_OFF     = H1_OFF   + (size_t)NN * NH * 4;  // NN*NC f32 (4 MB)
static constexpr size_t ZCAT_OFF   = HW_OFF   + (size_t)NN * NC * 4;  // 4 MB
static constexpr size_t Z_OFF      = ZCAT_OFF + (size_t)NN * NC * 4;  // NN*NZ f32 (2 MB)

// fp32 WMMA: D(16x16) = A(16x4) * B(4x16) + C   -- one tile per wave32
__device__ __forceinline__ v8f wmma_f32(v2f a, v2f b, v8f c) {
  return __builtin_amdgcn_wmma_f32_16x16x4_f32(
      /*neg_a=*/false, a, /*neg_b=*/false, b,
      /*c_mod=*/(short)0, c, /*reuse_a=*/false, /*reuse_b=*/false);
}

__device__ __forceinline__ float wave_reduce_add(float v) {
  #pragma unroll
  for (int off = 16; off > 0; off >>= 1) v += __shfl_xor(v, off, 32);
  return v;
}

__global__ void k_init(double* __restrict__ acc) {
  if (threadIdx.x < 2) acc[threadIdx.x] = 0.0;
}

// row_ptr[r] = lower_bound(edge_row, r) ; edge_row is sorted
__global__ void k_rowptr(const int* __restrict__ erow, int* __restrict__ rp) {
  int r = blockIdx.x * blockDim.x + threadIdx.x;
  if (r > NN) return;
  int lo = 0, hi = NE;
  while (lo < hi) {
    int mid = (lo + hi) >> 1;
    if (erow[mid] < r) lo = mid + 1; else hi = mid;
  }
  rp[r] = lo;
}

// C(NN x NH) = X(NN x NX) @ W1(NX x NH), fp32 WMMA, 1 tile per wave, 8 waves/block
__global__ void k_gemm_xw1(const float* __restrict__ A, const float* __restrict__ B,
                           float* __restrict__ C) {
  const int lane = threadIdx.x & 31;
  const int wave = threadIdx.x >> 5;
  const int w    = blockIdx.x * 8 + wave;
  const int tilesN = NH / 16;
  const int tm = (w / tilesN) * 16;
  const int tn = (w % tilesN) * 16;
  const int half = lane >> 4, lr = lane & 15;
  const float* arow = A + (size_t)(tm + lr) * NX + 2 * half;
  const float* bcol = B + tn + lr + (size_t)(2 * half) * NH;
  v8f c = {};
  #pragma unroll 8
  for (int k = 0; k < NX; k += 4) {
    v2f a, b;
    a.x = arow[k];              a.y = arow[k + 1];
    b.x = bcol[(size_t)k * NH]; b.y = bcol[(size_t)(k + 1) * NH];
    c = wmma_f32(a, b, c);
  }
  float* crow = C + (size_t)(tm + half * 8) * NH + tn + lr;
  #pragma unroll
  for (int i = 0; i < 8; ++i) crow[(size_t)i * NH] = c[i];
}

// C(NN x NC) = h1(NN x NH) @ [Wm | Ws]  (each NH x NZ)
__global__ void k_gemm_hw(const float* __restrict__ A, const float* __restrict__ Wm,
                          const float* __restrict__ Ws, float* __restrict__ C) {
  const int lane = threadIdx.x & 31;
  const int wave = threadIdx.x >> 5;
  const int w    = blockIdx.x * 8 + wave;
  const int tilesN = NC / 16;
  const int tm = (w / tilesN) * 16;
  const int tn = (w % tilesN) * 16;
  const int half = lane >> 4, lr = lane & 15;
  const int j = tn + lr;
  const float* bcol = ((j < NZ) ? (Wm + j) : (Ws + (j - NZ))) + (size_t)(2 * half) * NZ;
  const float* arow = A + (size_t)(tm + lr) * NH + 2 * half;
  v8f c = {};
  #pragma unroll 8
  for (int k = 0; k < NH; k += 4) {
    v2f a, b;
    a.x = arow[k];              a.y = arow[k + 1];
    b.x = bcol[(size_t)k * NZ]; b.y = bcol[(size_t)(k + 1) * NZ];
    c = wmma_f32(a, b, c);
  }
  float* crow = C + (size_t)(tm + half * 8) * NC + tn + lr;
  #pragma unroll
  for (int i = 0; i < 8; ++i) crow[(size_t)i * NC] = c[i];
}

// CSR SpMM: out[r,:] = sum_e vals[e] * H[col[e],:]  (one block per row, thread per col)
template <int F, bool RELU>
__global__ void k_spmm(const float* __restrict__ vals, const int* __restrict__ col,
                       const int* __restrict__ rp, const float* __restrict__ H,
                       float* __restrict__ out) {
  const int r = blockIdx.x;
  const int f = threadIdx.x;
  const int e0 = rp[r], e1 = rp[r + 1];
  float acc = 0.f;
  for (int e = e0; e < e1; ++e)
    acc = fmaf(vals[e], H[(size_t)col[e] * F + f], acc);
  if (RELU) acc = fmaxf(acc, 0.f);
  out[(size_t)r * F + f] = acc;
}

// Z = z_mean + eps*exp(z_ls); kl partial sum
__global__ void k_z_kl(const float* __restrict__ zcat, const float* __restrict__ eps,
                       float* __restrict__ Z, double* __restrict__ acc) {
  const int idx = blockIdx.x * blockDim.x + threadIdx.x; // < NN*NZ
  const int i = idx / NZ, f = idx % NZ;
  const float zm = zcat[(size_t)i * NC + f];
  const float zl = zcat[(size_t)i * NC + NZ + f];
  const float ez = expf(zl);
  Z[idx] = fmaf(eps[idx], ez, zm);
  float kl = 1.f + 2.f * zl - zm * zm - ez * ez;
  kl = wave_reduce_add(kl);
  __shared__ float s[8];
  const int lane = threadIdx.x & 31, wv = threadIdx.x >> 5;
  if (lane == 0) s[wv] = kl;
  __syncthreads();
  if (threadIdx.x == 0) {
    float t = 0.f;
    #pragma unroll
    for (int k = 0; k < 8; ++k) t += s[k];
    atomicAdd(&acc[1], (double)t);
  }
}

// A = Z @ Z^T (fp32 WMMA) fused with weighted-BCE partial sum.
// Each wave DMAs its 16x16 adj_label tile into LDS via the Tensor Data Mover
// while the WMMA chain runs, then waits on TENSORcnt and reads labels from LDS.
// A is written with non-temporal stores (streamed once, never re-read).
__global__ void k_zzt_loss(const float* __restrict__ Z, const float* __restrict__ lab,
                           float* __restrict__ A, double* __restrict__ acc) {
  // ONLY shared allocation in this kernel -> static LDS base offset 0.
  __shared__ float lsm[8 * 256];  // 1 KB label tile per wave (8 waves)

  const int lane = threadIdx.x & 31;
  const int wave = threadIdx.x >> 5;
  const int w    = blockIdx.x * 8 + wave;
  const int tilesN = NN / 16;
  const int tm = (w / tilesN) * 16;
  const int tn = (w % tilesN) * 16;
  const int half = lane >> 4, lr = lane & 15;

  // ---- issue TDM: 16x16 f32 tile of adj_label -> LDS at byte offset wave*1024 ----
  const unsigned long long gaddr =
      (unsigned long long)(const void*)(lab + (size_t)tm * NN + tn);
  u32x4 g0;
  g0.x = 1u;                                    // count=1, user mode, no gather
  g0.y = (u32)(wave * 1024);                    // lds_addr (bytes)
  g0.z = (u32)(gaddr & 0xFFFFFFFFull);          // global_addr[31:0]
  g0.w = (u32)((gaddr >> 32) & 0x1FFFFFFull)    // global_addr[56:32]
         | (2u << 30);                          // type=2 ("image")
  i32x8 g1;
  g1[0] = 0x20000;                    // workgroup_mask=0, data_size=2 (4 bytes)
  g1[1] = (int)(8192u << 16);         // tensor_dim0[15:0]=8192 in [31:16]
  g1[2] = (int)(8192u << 16);         // tensor_dim0[31:16]=0 ; tensor_dim1[15:0]=8192
  g1[3] = (int)(16u << 16);           // tensor_dim1[31:16]=0 ; tile_dim0=16
  g1[4] = 16;                         // tile_dim1=16 ; tile_dim2=0
  g1[5] = 8192;                       // tensor_dim0_stride[31:0] (elements)
  g1[6] = 0;                          // stride hi / dim1_stride lo
  g1[7] = 0;
  i32x4 gz4 = {0, 0, 0, 0};                    // 2-D tensor: groups 2/3 unused
  i32x8 gz8 = {0, 0, 0, 0, 0, 0, 0, 0};        // trailing group (6-arg variant)
  __builtin_amdgcn_tensor_load_to_lds(g0, g1, gz4, gz4, gz8, 0);

  // ---- WMMA chain overlaps the TDM transfer ----
  const float* zr = Z + (size_t)(tm + lr) * NZ + 2 * half; // A frag: row of Z
  const float* zc = Z + (size_t)(tn + lr) * NZ + 2 * half; // B frag: Z row (col of Z^T)
  v8f c = {};
  #pragma unroll
  for (int k = 0; k < NZ; k += 4) {
    v2f a, b;
    a.x = zr[k]; a.y = zr[k + 1];
    b.x = zc[k]; b.y = zc[k + 1];
    c = wmma_f32(a, b, c);
  }

  __builtin_amdgcn_s_wait_tensorcnt(0);
  __asm__ __volatile__("" ::: "memory");   // keep LDS reads after the wait

  const float* lt = &lsm[wave * 256];      // row-major 16x16 tile
  float wce = 0.f;
  #pragma unroll
  for (int i = 0; i < 8; ++i) {
    const int row = half * 8 + i;
    const size_t off = (size_t)(tm + row) * NN + tn + lr;
    const float x = c[i];
    const float z = lt[row * 16 + lr];
    // softplus(-x) = max(-x,0) + log1p(exp(-|x|))
    const float sp = fmaxf(-x, 0.f) + log1pf(expf(-fabsf(x)));
    wce += (1.f - z) * x + fmaf(POS_W - 1.f, z, 1.f) * sp;
    __builtin_nontemporal_store(x, &A[off]);
  }

  wce = wave_reduce_add(wce);
  // reuse lsm slot [wave*256] for the block reduction (own tile fully consumed)
  if (lane == 0) lsm[wave * 256] = wce;
  __syncthreads();
  if (threadIdx.x == 0) {
    float t = 0.f;
    #pragma unroll
    for (int k = 0; k < 8; ++k) t += lsm[k * 256];
    atomicAdd(&acc[0], (double)t);
  }
}

__global__ void k_finalize(const double* __restrict__ acc, float* __restrict__ loss_out) {
  if (threadIdx.x == 0 && blockIdx.x == 0) {
    const double lrec = (double)NORM_LOSS * (acc[0] / N2d);
    const double llat = -0.5 * acc[1] / N2d;
    loss_out[0] = (float)(lrec + llat);
  }
}

extern "C" void kernel_launch(void* const* d_in, const int* in_sizes, int n_in,
                              void* d_out, int out_size, void* d_ws, size_t ws_size,
                              hipStream_t stream) {
  const float* X    = (const float*)d_in[0];
  const float* W1   = (const float*)d_in[1];
  const float* Wm   = (const float*)d_in[2];
  const float* Ws   = (const float*)d_in[3];
  const float* vals = (const float*)d_in[4];
  const float* lab  = (const float*)d_in[5];
  const float* eps  = (const float*)d_in[6];
  const int*   erow = (const int*)d_in[7];
  const int*   ecol = (const int*)d_in[8];

  char*   ws   = (char*)d_ws;
  double* acc  = (double*)(ws + ACC_OFF);
  int*    rp   = (int*)(ws + ROWPTR_OFF);
  float*  XW1  = (float*)(ws + XW1_OFF);
  float*  H1   = (float*)(ws + H1_OFF);
  float*  HW   = (float*)(ws + HW_OFF);
  float*  ZCAT = (float*)(ws + ZCAT_OFF);
  float*  Z    = (float*)(ws + Z_OFF);

  float* A    = (float*)d_out;
  float* loss = A + (size_t)NN * NN;

  k_init<<<1, 64, 0, stream>>>(acc);
  k_rowptr<<<(NN + 1 + 255) / 256, 256, 0, stream>>>(erow, rp);
  k_gemm_xw1<<<(NN / 16) * (NH / 16) / 8, 256, 0, stream>>>(X, W1, XW1);
  k_spmm<NH, true><<<NN, NH, 0, stream>>>(vals, ecol, rp, XW1, H1);
  k_gemm_hw<<<(NN / 16) * (NC / 16) / 8, 256, 0, stream>>>(H1, Wm, Ws, HW);
  k_spmm<NC, false><<<NN, NC, 0, stream>>>(vals, ecol, rp, HW, ZCAT);
  k_z_kl<<<(NN * NZ) / 256, 256, 0, stream>>>(ZCAT, eps, Z, acc);
  k_zzt_loss<<<(NN / 16) * (NN / 16) / 8, 256, 0, stream>>>(Z, lab, A, acc);
  k_finalize<<<1, 1, 0, stream>>>(acc, loss);
}